// SymmetricChannel_6296422056028
// MI455X (gfx1250) — compile-verified
//
#include <hip/hip_runtime.h>
#include <stdint.h>

// Problem constants from the reference: B=512, L=32, V=2048, P=0.1
#define ROWS      16384        // B*L
#define VDIM      2048
#define NOISE_W   2047         // V-1
#define NELEM     33554432UL   // ROWS*VDIM
#define WAVES_PB  8

#define PROB        0.1f
#define INV_VM2     (1.0f / 2046.0f)            // 1/(V-2)
#define SCALE_T     (1.0f + 1.0f / 2046.0f)     // 1 + 1/(V-2)
#define LOG_KEEP    (-0.10536051565782628f)     // ln(0.9)
#define LOG_REP     (-9.9262261162834283f)      // ln(0.1/2046)

typedef float v4f __attribute__((ext_vector_type(4)));

// Symmetric-channel log-prob mixing: logaddexp(lv + ln(1-P), ln(p_tail) + ln(P/(V-2)))
__device__ __forceinline__ float chan_mix(float lv, float e0) {
    float pt = 1.0f - expf(lv) - e0;
    pt = fminf(fmaxf(pt, 0.0f), 1.0f);
    const float a  = lv + LOG_KEEP;
    const float b  = logf(pt) + LOG_REP;        // log(0) -> -inf collapses to keep-branch
    const float mx = fmaxf(a, b);
    const float mn = fminf(a, b);
    return mx + log1pf(expf(mn - mx));
}

__global__ __launch_bounds__(256) void symchan_kernel(
    const float* __restrict__ messages,
    const float* __restrict__ logits,
    const float* __restrict__ noise,
    float* __restrict__ out)
{
    __shared__ float sm[WAVES_PB][VDIM];   // 64 KB: one staged message row per wave

    const int wave = threadIdx.x >> 5;
    const int lane = threadIdx.x & 31;
    const size_t row = (size_t)blockIdx.x * WAVES_PB + wave;

    const float* __restrict__ mrow = messages + row * VDIM;
    const float* __restrict__ lrow = logits   + row * VDIM;
    const float* __restrict__ nrow = noise    + row * (size_t)NOISE_W;
    float* __restrict__ sm_row = &sm[wave][0];

    // ---- Issue the 8KB message-row DMA into LDS (CDNA5 async global->LDS).
    // Each b128 instruction moves 32 lanes * 16B = 512B; 16 instructions = 8KB.
    // We do NOT wait yet: the logits pass below is independent of LDS and
    // executes under the in-flight ASYNCcnt, hiding the DMA latency.
    {
        const uint32_t lds_base = (uint32_t)(uintptr_t)sm_row;
        const char* gbase = (const char*)mrow;
        #pragma unroll
        for (int j = 0; j < 16; ++j) {
            const uint32_t loff = lds_base + (uint32_t)(lane * 16 + j * 512);
            const void* gaddr = gbase + (lane * 16 + j * 512);
            asm volatile("global_load_async_to_lds_b128 %0, %1, off"
                         :: "v"(loff), "v"(gaddr) : "memory");
        }
    }

    // ---- Pass A (overlapped with DMA): symmetric-channel logit mixing + logits
    // passthrough. Pure streaming: b128 NT loads/stores, no LDS dependence.
    {
        float* __restrict__ oln = out + 1UL * NELEM + row * VDIM;      // logits_noisy
        float* __restrict__ olc = out + 3UL * NELEM + row * VDIM;      // logits copy
        const float e0 = expf(lrow[0]);                                // broadcast load
        #pragma unroll 2
        for (int k = 0; k < 16; ++k) {
            const int j0 = (lane + (k << 5)) << 2;
            const v4f l4 = __builtin_nontemporal_load((const v4f*)(lrow + j0));
            v4f r;
            r.x = (j0 == 0) ? l4.x : chan_mix(l4.x, e0);               // EOS column copied
            r.y = chan_mix(l4.y, e0);
            r.z = chan_mix(l4.z, e0);
            r.w = chan_mix(l4.w, e0);
            __builtin_nontemporal_store(r,  (v4f*)(oln + j0));
            __builtin_nontemporal_store(l4, (v4f*)(olc + j0));
        }
    }

    // ---- Drain this wave's ASYNCcnt before touching the staged LDS row.
    asm volatile("s_wait_asynccnt 0" ::: "memory");

    // ---- Pass B: masked row-sum of t over noise columns 0..V-2 (wave32 reduction).
    // Chunk-of-4 mapping: lane owns columns [4*(lane+32k), +4). Noise row is only
    // 4B-aligned (stride 2047) -> scalar noise loads; message data from LDS b128.
    float sum = 0.0f;
    #pragma unroll 4
    for (int k = 0; k < 16; ++k) {
        const int c0 = (lane + (k << 5)) << 2;        // 0..2044 step 4
        const v4f m4 = *(const v4f*)&sm_row[c0];
        if (nrow[c0 + 0] < PROB) sum += m4.x;
        if (nrow[c0 + 1] < PROB) sum += m4.y;
        if (nrow[c0 + 2] < PROB) sum += m4.z;
        if (c0 + 3 < NOISE_W) {                        // excludes only c == 2047
            if (nrow[c0 + 3] < PROB) sum += m4.w;
        }
    }
    #pragma unroll
    for (int off = 16; off > 0; off >>= 1)
        sum += __shfl_xor(sum, off, 32);
    const float addc = sum * INV_VM2;

    // ---- Pass C: m_noisy + messages passthrough. Shift m[j-1]/noise[j-1] resolves
    // inside the float4 except element 0 of each chunk (LDS / WGP$-hit noise).
    {
        float* __restrict__ omn = out + row * VDIM;                    // m_noisy
        float* __restrict__ omc = out + 2UL * NELEM + row * VDIM;      // messages copy
        #pragma unroll 4
        for (int k = 0; k < 16; ++k) {
            const int j0 = (lane + (k << 5)) << 2;
            const v4f m4 = *(const v4f*)&sm_row[j0];
            v4f r;
            if (j0 == 0) {
                r.x = m4.x;                                            // EOS column copied
            } else {
                const float t = (nrow[j0 - 1] < PROB) ? sm_row[j0 - 1] : 0.0f;
                r.x = m4.x + addc - t * SCALE_T;
            }
            { const float t = (nrow[j0 + 0] < PROB) ? m4.x : 0.0f; r.y = m4.y + addc - t * SCALE_T; }
            { const float t = (nrow[j0 + 1] < PROB) ? m4.y : 0.0f; r.z = m4.z + addc - t * SCALE_T; }
            { const float t = (nrow[j0 + 2] < PROB) ? m4.z : 0.0f; r.w = m4.w + addc - t * SCALE_T; }
            __builtin_nontemporal_store(r,  (v4f*)(omn + j0));         // streamed, never re-read
            __builtin_nontemporal_store(m4, (v4f*)(omc + j0));
        }
    }
}

extern "C" void kernel_launch(void* const* d_in, const int* in_sizes, int n_in,
                              void* d_out, int out_size, void* d_ws, size_t ws_size,
                              hipStream_t stream) {
    (void)in_sizes; (void)n_in; (void)out_size; (void)d_ws; (void)ws_size;
    const float* messages = (const float*)d_in[0];
    const float* logits   = (const float*)d_in[1];
    const float* noise    = (const float*)d_in[2];
    float* out = (float*)d_out;

    const dim3 grid(ROWS / WAVES_PB);   // 2048 blocks, one wave32 per row
    const dim3 block(32 * WAVES_PB);    // 256 threads = 8 wave32
    symchan_kernel<<<grid, block, 0, stream>>>(messages, logits, noise, out);
}